// GRU_83554293777061
// MI455X (gfx1250) — compile-verified
//
#include <hip/hip_runtime.h>
#include <hip/hip_bf16.h>
#include <math.h>

typedef __attribute__((ext_vector_type(2))) float v2f;
typedef __attribute__((ext_vector_type(8))) float v8f;

#define Bsz 32
#define Tt  512
#define Hh  512
#define NG  1536   // 3*H
#define EPSV 1e-5f

#define USE_ASYNC_LDS 1

#define SB_PITCH 516                    // 512 + 4 pad -> bank stride 4 (mod 64), conflict-free
#define SB_FLOATS (48 * SB_PITCH)       // 3 gate tiles x 16 rows
#define HH_FLOATS (3 * Bsz * 16)
#define SMEM_RC_BYTES ((SB_FLOATS + HH_FLOATS) * 4)

// ---------------------------------------------------------------------------
// Device-wide barrier (32 persistent blocks, all co-resident).
// ---------------------------------------------------------------------------
__device__ __forceinline__ void grid_barrier(int* cnt, int* gen, int nblk) {
    __threadfence();
    __syncthreads();
    if (threadIdx.x == 0) {
        int g = __hip_atomic_load(gen, __ATOMIC_ACQUIRE, __HIP_MEMORY_SCOPE_AGENT);
        int prev = __hip_atomic_fetch_add(cnt, 1, __ATOMIC_ACQ_REL, __HIP_MEMORY_SCOPE_AGENT);
        if (prev == nblk - 1) {
            __hip_atomic_store(cnt, 0, __ATOMIC_RELAXED, __HIP_MEMORY_SCOPE_AGENT);
            __hip_atomic_fetch_add(gen, 1, __ATOMIC_RELEASE, __HIP_MEMORY_SCOPE_AGENT);
        } else {
            while (__hip_atomic_load(gen, __ATOMIC_ACQUIRE, __HIP_MEMORY_SCOPE_AGENT) == g) {
                __builtin_amdgcn_s_sleep(1);
            }
        }
    }
    __syncthreads();
}

// ---------------------------------------------------------------------------
// Phase A: XI = X . Wih^T + bih   (M=16384, N=1536, K=512), fp32 WMMA.
// 2x2 register blocking: each wave computes a 32x32 output block (4 tiles)
// from 2 A-frags + 2 B-frags per K=4 step -> 1 load : 1 wmma.
// grid = (48, 128), block = 128 (4 waves).
// ---------------------------------------------------------------------------
__global__ void __launch_bounds__(128)
gru_xi_gemm(const float* __restrict__ X, const float* __restrict__ Wih,
            const float* __restrict__ bih, float* __restrict__ XI) {
    const int K = Hh;
    const int lane = threadIdx.x & 31;
    const int wave = threadIdx.x >> 5;
    const int nt0 = blockIdx.x * 2;                    // 0..94
    const int mt0 = (blockIdx.y * 4 + wave) * 2;       // 0..1022
    const int m  = lane & 15;
    const int n  = lane & 15;
    const int kb = (lane >> 4) << 1;                   // 0 | 2

    const float* a0p = X + (size_t)(mt0 * 16 + m) * K + kb;
    const float* a1p = a0p + (size_t)16 * K;
    const float* b0p = Wih + (size_t)(nt0 * 16 + n) * K + kb;
    const float* b1p = b0p + (size_t)16 * K;

    v8f c00 = {}, c01 = {}, c10 = {}, c11 = {};
    #pragma unroll 4
    for (int k = 0; k < K; k += 4) {
        v2f a0 = *(const v2f*)(a0p + k);
        v2f a1 = *(const v2f*)(a1p + k);
        v2f b0 = *(const v2f*)(b0p + k);
        v2f b1 = *(const v2f*)(b1p + k);
        c00 = __builtin_amdgcn_wmma_f32_16x16x4_f32(false, a0, false, b0, (short)0, c00, false, false);
        c01 = __builtin_amdgcn_wmma_f32_16x16x4_f32(false, a0, false, b1, (short)0, c01, false, false);
        c10 = __builtin_amdgcn_wmma_f32_16x16x4_f32(false, a1, false, b0, (short)0, c10, false, false);
        c11 = __builtin_amdgcn_wmma_f32_16x16x4_f32(false, a1, false, b1, (short)0, c11, false, false);
    }

    const int nA = nt0 * 16 + n;
    const int nB = nA + 16;
    const float biasA = bih[nA];
    const float biasB = bih[nB];
    // D layout: VGPR r holds M=r (lanes 0-15) / M=8+r (lanes 16-31)
    const int mrow0 = mt0 * 16 + ((lane >> 4) << 3);
    float* o = XI + (size_t)mrow0 * NG;
    #pragma unroll
    for (int r = 0; r < 8; ++r) {
        o[(size_t)r * NG + nA]        = c00[r] + biasA;
        o[(size_t)r * NG + nB]        = c01[r] + biasB;
        o[(size_t)(r + 16) * NG + nA] = c10[r] + biasA;
        o[(size_t)(r + 16) * NG + nB] = c11[r] + biasB;
    }
}

// ---------------------------------------------------------------------------
// Phase B: persistent recurrence.  32 blocks (one per hidden 16-col tile),
// 192 threads = 6 waves = (gate r/z/n) x (batch tile 0/1).
// Whh tiles staged once in LDS (async-to-LDS engine), padded pitch 516 so
// the per-step ds_load_b64 B-frag reads are bank-conflict-free.
// ---------------------------------------------------------------------------
__global__ void __launch_bounds__(192)
gru_recurrence(const float* __restrict__ XI, const float* __restrict__ Whh,
               const float* __restrict__ bhh, float* __restrict__ Y,
               float* __restrict__ H0, float* __restrict__ H1,
               int* __restrict__ bar_cnt, int* __restrict__ bar_gen) {
    extern __shared__ float smem[];
    float* sB = smem;                       // [48][SB_PITCH] row index = g*16 + r
    float* hh = smem + SB_FLOATS;           // [3][32][16]
    #define HHE(g_, b_, jl_) hh[(((g_) * Bsz) + (b_)) * 16 + (jl_)]

    const int K = Hh;
    const int tid  = threadIdx.x;
    const int lane = tid & 31;
    const int wave = tid >> 5;
    const int mt   = wave & 1;
    const int g    = wave >> 1;            // 0=r, 1=z, 2=n
    const int jt   = blockIdx.x;           // hidden tile 0..31
    const int nblk = gridDim.x;

    const int m  = lane & 15;
    const int n  = lane & 15;
    const int kb = (lane >> 4) << 1;
    const float bias = bhh[(g * 32 + jt) * 16 + n];

    // ---- stage this block's 48 Whh rows (3 gate tiles x 16) into LDS ----
    {
        const int tot = 48 * (Hh / 4);                     // float4 transfers
        for (int e = tid; e < tot; e += blockDim.x) {
            const int row = e / (Hh / 4);                  // 0..47 = g*16 + r
            const int gg  = row >> 4;
            const int rr  = row & 15;
            const int c4  = e % (Hh / 4);
            const float* src = Whh + (size_t)((gg * 32 + jt) * 16 + rr) * Hh + c4 * 4;
            float* dstp = &sB[row * SB_PITCH + c4 * 4];
#if USE_ASYNC_LDS
            unsigned int dst = (unsigned int)(uintptr_t)dstp;   // low 32 bits = LDS offset
            asm volatile("global_load_async_to_lds_b128 %0, %1, off"
                         :: "v"(dst), "v"(src) : "memory");
#else
            *(float4*)dstp = *(const float4*)src;
#endif
        }
#if USE_ASYNC_LDS
#if __has_builtin(__builtin_amdgcn_s_wait_asynccnt)
        __builtin_amdgcn_s_wait_asynccnt(0);
#else
        asm volatile("s_wait_asynccnt 0" ::: "memory");
#endif
#endif
    }

    // ---- h(0) = 0 for this block's 16 columns ----
    for (int e = tid; e < Bsz * 16; e += blockDim.x)
        H0[(e >> 4) * K + jt * 16 + (e & 15)] = 0.0f;
    grid_barrier(bar_cnt, bar_gen, nblk);   // also makes staged LDS visible block-wide

    const float* bsrow = &sB[(g * 16 + n) * SB_PITCH + kb];

    float* Hold = H0;
    float* Hnew = H1;
    for (int t = 0; t < Tt; ++t) {
        // ---- HH tile = Hold(16x512) . Whh_tile^T(512x16), B from LDS ----
        const float* arow = Hold + (size_t)(mt * 16 + m) * K + kb;
        v8f c = {};
        #pragma unroll 8
        for (int k = 0; k < K; k += 4) {
            v2f a = *(const v2f*)(arow + k);
            v2f b = *(const v2f*)(bsrow + k);
            c = __builtin_amdgcn_wmma_f32_16x16x4_f32(false, a, false, b,
                                                      (short)0, c, false, false);
        }
        const int mrow0 = mt * 16 + ((lane >> 4) << 3);
        #pragma unroll
        for (int r = 0; r < 8; ++r) HHE(g, mrow0 + r, n) = c[r] + bias;
        __syncthreads();

        // ---- gate combine + state update (32 x 16 elements) ----
        for (int e = tid; e < Bsz * 16; e += blockDim.x) {
            const int b  = e >> 4;
            const int jl = e & 15;
            const int j  = jt * 16 + jl;
            const float* xi = XI + ((size_t)b * Tt + t) * NG;
            const float hr = HHE(0, b, jl);
            const float hz = HHE(1, b, jl);
            const float hn = HHE(2, b, jl);
            const float rr = 1.0f / (1.0f + __expf(-(xi[j]      + hr)));
            const float zz = 1.0f / (1.0f + __expf(-(xi[Hh + j] + hz)));
            const float nn = tanhf(xi[2 * Hh + j] + rr * hn);
            const float h  = Hold[b * K + j];
            const float ho = (1.0f - zz) * nn + zz * h;
            Hnew[b * K + j] = ho;
            Y[((size_t)b * Tt + t) * Hh + j] = ho;
        }
        grid_barrier(bar_cnt, bar_gen, nblk);   // publishes Hnew, guards LDS reuse
        float* tmp = Hold; Hold = Hnew; Hnew = tmp;
    }
    #undef HHE
}

// ---------------------------------------------------------------------------
// Mask (t >= seq_len -> 0) + LayerNorm over H.  One block per (b,t) row.
// ---------------------------------------------------------------------------
__global__ void __launch_bounds__(256)
gru_mask_ln(const float* __restrict__ X, const float* __restrict__ gamma,
            const float* __restrict__ beta, const int* __restrict__ seq_lens,
            float* __restrict__ Out) {
    __shared__ float s1[256];
    __shared__ float s2[256];
    const int bt = blockIdx.x;
    const int b  = bt >> 9;
    const int t  = bt & (Tt - 1);
    const bool valid = t < seq_lens[b];
    const float* row = X + (size_t)bt * Hh;

    const int i0 = threadIdx.x;
    const int i1 = threadIdx.x + 256;
    float v0 = valid ? row[i0] : 0.0f;
    float v1 = valid ? row[i1] : 0.0f;

    s1[threadIdx.x] = v0 + v1;
    s2[threadIdx.x] = v0 * v0 + v1 * v1;
    __syncthreads();
    for (int s = 128; s > 0; s >>= 1) {
        if (threadIdx.x < s) {
            s1[threadIdx.x] += s1[threadIdx.x + s];
            s2[threadIdx.x] += s2[threadIdx.x + s];
        }
        __syncthreads();
    }
    const float mu  = s1[0] * (1.0f / Hh);
    const float var = s2[0] * (1.0f / Hh) - mu * mu;
    const float rs  = rsqrtf(var + EPSV);

    float* orow = Out + (size_t)bt * Hh;
    orow[i0] = (v0 - mu) * rs * gamma[i0] + beta[i0];
    orow[i1] = (v1 - mu) * rs * gamma[i1] + beta[i1];
}

// ---------------------------------------------------------------------------
extern "C" void kernel_launch(void* const* d_in, const int* in_sizes, int n_in,
                              void* d_out, int out_size, void* d_ws, size_t ws_size,
                              hipStream_t stream) {
    const float* x     = (const float*)d_in[0];
    const float* Wih   = (const float*)d_in[1];   // [4, 1536, 512]
    const float* Whh   = (const float*)d_in[2];   // [4, 1536, 512]
    const float* bih   = (const float*)d_in[3];   // [4, 1536]
    const float* bhh   = (const float*)d_in[4];   // [4, 1536]
    const float* gamma = (const float*)d_in[5];   // [2, 512]
    const float* beta  = (const float*)d_in[6];   // [2, 512]
    const int*   lens  = (const int*)d_in[7];     // [32]

    char* ws = (char*)d_ws;
    int*   bar  = (int*)ws;
    float* XI   = (float*)(ws + 256);
    float* bufA = XI   + (size_t)Bsz * Tt * NG;
    float* bufB = bufA + (size_t)Bsz * Tt * Hh;
    float* H0   = bufB + (size_t)Bsz * Tt * Hh;
    float* H1   = H0   + (size_t)Bsz * Hh;

    hipMemsetAsync(bar, 0, 2 * sizeof(int), stream);

    const dim3 gridXI(NG / 32, (Bsz * Tt) / 128);        // (48, 128)
    const dim3 blkXI(128);
    const dim3 gridRC(Hh / 16);                          // 32 persistent blocks
    const dim3 blkRC(192);
    const dim3 gridLN(Bsz * Tt);
    const dim3 blkLN(256);

    const size_t wstride = (size_t)NG * Hh;
    const size_t bstride = (size_t)NG;

    const float* cur = x;
    float* dst[4] = { bufA, bufB, bufB, bufA };

    int idx = 0;
    for (int l = 0; l < 2; ++l) {
        for (int k = 0; k < 2; ++k, ++idx) {
            gru_xi_gemm<<<gridXI, blkXI, 0, stream>>>(
                cur, Wih + idx * wstride, bih + idx * bstride, XI);
            gru_recurrence<<<gridRC, blkRC, SMEM_RC_BYTES, stream>>>(
                XI, Whh + idx * wstride, bhh + idx * bstride,
                dst[idx], H0, H1, bar, bar + 1);
            cur = dst[idx];
        }
        float* lnOut = (l == 0) ? bufA : (float*)d_out;
        gru_mask_ln<<<gridLN, blkLN, 0, stream>>>(
            cur, gamma + l * Hh, beta + l * Hh, lens, lnOut);
        cur = lnOut;
    }
}